// Warper_42442866819760
// MI455X (gfx1250) — compile-verified
//
#include <hip/hip_runtime.h>
#include <math.h>
#include <stdint.h>

// Warper: 8-tap windowed-sinc flow warp, 3x720x1280 f32.
// Gather/L0-bound; WMMA inapplicable (per-lane private 8x8 stencils).
// CDNA5 paths: global_load_async_to_lds_b32 + s_wait_asynccnt (flow staging),
// global_prefetch_b8 (footprint warm), b128 gathers with 24-bit imm-offset
// folding across channel planes, wave32 32-wide row blocking.

namespace {
constexpr int kC = 3;
constexpr int kH = 720;
constexpr int kW = 1280;
constexpr int kHW = kH * kW;
constexpr float kPI = 3.14159265358979323846f;
}

typedef float f4 __attribute__((ext_vector_type(4)));
typedef f4 uf4 __attribute__((aligned(4)));   // dword-aligned b128 loads are legal on gfx1250

__device__ __forceinline__ float hdot8(uf4 a, uf4 b, const float cx[8]) {
    float s = a.x * cx[0];
    s = fmaf(a.y, cx[1], s);
    s = fmaf(a.z, cx[2], s);
    s = fmaf(a.w, cx[3], s);
    s = fmaf(b.x, cx[4], s);
    s = fmaf(b.y, cx[5], s);
    s = fmaf(b.z, cx[6], s);
    s = fmaf(b.w, cx[7], s);
    return s;
}

// offsets o_j = j-3, j=0..7 -> {-3..4}
// coef_j = cos(pi*(s-o_j)/8) * sinc(s-o_j)
//   cos(pi*(s-o)/8) = ca*cos(pi*o/8) + sa*sin(pi*o/8)      (angle addition)
//   sinc(s-o)       = (-1)^o * sin(pi*s) / (pi*(s-o)),  t==0 -> 1
// -> only 3 native transcendentals per axis + 8 v_rcp_f32.
__device__ __forceinline__ void sinc_coeffs(float s, float coef[8]) {
    const float CB[8] = {0.3826834324f, 0.7071067812f, 0.9238795325f, 1.0f,
                         0.9238795325f, 0.7071067812f, 0.3826834324f, 0.0f};
    const float SB[8] = {-0.9238795325f, -0.7071067812f, -0.3826834324f, 0.0f,
                         0.3826834324f, 0.7071067812f, 0.9238795325f, 1.0f};
    const float SG[8] = {-1.f, 1.f, -1.f, 1.f, -1.f, 1.f, -1.f, 1.f};
    float sp = __sinf(kPI * s);            // v_sin_f32 path; arg in [0, pi)
    float sa, ca;
    __sincosf((kPI / 8.0f) * s, &sa, &ca); // v_sin/v_cos; arg in [0, pi/8)
#pragma unroll
    for (int j = 0; j < 8; ++j) {
        float t = s - (float)(j - 3);
        float c8 = fmaf(ca, CB[j], sa * SB[j]);
        float v = c8 * SG[j] * sp * __builtin_amdgcn_rcpf(kPI * t);
        coef[j] = (t == 0.0f) ? 1.0f : v;
    }
}

__launch_bounds__(256)
__global__ void warp_sinc_kernel(const float* __restrict__ xin,
                                 const float* __restrict__ flow,
                                 float* __restrict__ out) {
    __shared__ float sflow[2][256];
    const int tid = threadIdx.y * 32 + threadIdx.x;
    const int w = blockIdx.x * 32 + threadIdx.x;
    const int h = blockIdx.y * 8 + threadIdx.y;
    const int idx = h * kW + w;

    // ---- stage this lane's flow through LDS via the CDNA5 async pipe ----
    uint32_t lds0 = (uint32_t)(uintptr_t)&sflow[0][tid];
    uint32_t lds1 = (uint32_t)(uintptr_t)&sflow[1][tid];
    uint64_t gfx = (uint64_t)(uintptr_t)(flow + idx);
    uint64_t gfy = (uint64_t)(uintptr_t)(flow + kHW + idx);
    asm volatile("global_load_async_to_lds_b32 %0, %1, off" :: "v"(lds0), "v"(gfx) : "memory");
    asm volatile("global_load_async_to_lds_b32 %0, %1, off" :: "v"(lds1), "v"(gfy) : "memory");
    asm volatile("s_wait_asynccnt 0" ::: "memory");
    const float fx = sflow[0][tid];
    const float fy = sflow[1][tid];

    const float rfx = floorf(fx);
    const float rfy = floorf(fy);
    const float sx = fx - rfx;
    const float sy = fy - rfy;
    const int bx = w + (int)rfx - 3;   // first x tap (unclamped)
    const int by = h + (int)rfy - 3;   // first y tap (unclamped)

    // clamped row offsets (y clamping never breaks x contiguity)
    int roff[8];
#pragma unroll
    for (int i = 0; i < 8; ++i) {
        int iy = by + i;
        iy = iy < 0 ? 0 : (iy > kH - 1 ? kH - 1 : iy);
        roff[i] = iy * kW;
    }

    // warm the channel-0 footprint rows while coefficient math runs;
    // channels 1/2 ride the load pipeline (same rows, +kHW / +2*kHW).
    const int bxs = bx < 0 ? 0 : (bx > kW - 8 ? kW - 8 : bx);
#pragma unroll
    for (int i = 0; i < 8; ++i)
        __builtin_prefetch(xin + roff[i] + bxs, 0, 3);

    float cx[8], cy[8];
    sinc_coeffs(sx, cx);
    sinc_coeffs(sy, cy);

    float acc0 = 0.f, acc1 = 0.f, acc2 = 0.f;

    const bool fastx = (bx >= 0) && (bx <= kW - 8);
    if (fastx) {
        // interior: one base address per row; 6 loads with folded 24-bit
        // immediate offsets {0,16, +kHW*4, +kHW*4+16, +2kHW*4, +2kHW*4+16}
#pragma unroll
        for (int i = 0; i < 8; ++i) {
            const float* rp = xin + (roff[i] + bx);
            uf4 a0 = *(const uf4*)(rp);
            uf4 b0 = *(const uf4*)(rp + 4);
            uf4 a1 = *(const uf4*)(rp + kHW);
            uf4 b1 = *(const uf4*)(rp + kHW + 4);
            uf4 a2 = *(const uf4*)(rp + 2 * kHW);
            uf4 b2 = *(const uf4*)(rp + 2 * kHW + 4);
            acc0 = fmaf(cy[i], hdot8(a0, b0, cx), acc0);
            acc1 = fmaf(cy[i], hdot8(a1, b1, cx), acc1);
            acc2 = fmaf(cy[i], hdot8(a2, b2, cx), acc2);
        }
    } else {
        // x-edge: per-tap clamped gathers, one address + 3 imm-offset loads
        int ix[8];
#pragma unroll
        for (int j = 0; j < 8; ++j) {
            int t = bx + j;
            ix[j] = t < 0 ? 0 : (t > kW - 1 ? kW - 1 : t);
        }
#pragma unroll
        for (int i = 0; i < 8; ++i) {
            float s0 = 0.f, s1 = 0.f, s2 = 0.f;
#pragma unroll
            for (int j = 0; j < 8; ++j) {
                const float* tp = xin + (roff[i] + ix[j]);
                s0 = fmaf(tp[0], cx[j], s0);
                s1 = fmaf(tp[kHW], cx[j], s1);
                s2 = fmaf(tp[2 * kHW], cx[j], s2);
            }
            acc0 = fmaf(cy[i], s0, acc0);
            acc1 = fmaf(cy[i], s1, acc1);
            acc2 = fmaf(cy[i], s2, acc2);
        }
    }

    float* op = out + idx;
    op[0] = acc0;
    op[kHW] = acc1;
    op[2 * kHW] = acc2;
}

extern "C" void kernel_launch(void* const* d_in, const int* in_sizes, int n_in,
                              void* d_out, int out_size, void* d_ws, size_t ws_size,
                              hipStream_t stream) {
    const float* x = (const float*)d_in[0];     // (1,3,720,1280) f32
    const float* flow = (const float*)d_in[1];  // (1,2,720,1280) f32
    float* out = (float*)d_out;                 // (1,3,720,1280) f32
    (void)in_sizes; (void)n_in; (void)out_size; (void)d_ws; (void)ws_size;
    dim3 block(32, 8);                 // one wave32 per 32-pixel row segment
    dim3 grid(kW / 32, kH / 8);        // 40 x 90, exact cover
    warp_sinc_kernel<<<grid, block, 0, stream>>>(x, flow, out);
}